// ScaledDotProductAttention_65343632441709
// MI455X (gfx1250) — compile-verified
//
#include <hip/hip_runtime.h>
#include <math.h>

typedef __attribute__((ext_vector_type(2))) float v2f;
typedef __attribute__((ext_vector_type(4))) float v4f;
typedef __attribute__((ext_vector_type(8))) float v8f;
typedef __attribute__((ext_vector_type(4))) unsigned int u32x4;
typedef __attribute__((ext_vector_type(8))) int i32x8;
typedef __attribute__((ext_vector_type(4))) int i32x4;

constexpr int BB = 64;    // batch * heads
constexpr int LL = 1024;  // sequence length
constexpr int DD = 64;    // head dim
constexpr float INV_T = 0.125f;  // 1 / sqrt(64)

// LDS row strides (in floats): 64 data + 4 pad = 68 ≡ 4 (mod 64 banks).
// The TDM descriptor reproduces this via pad_interval=64dw, pad_amount=4dw.
constexpr int QS = 68;
constexpr int SS = 1028;  // score buffer row stride

// ---- TDM: DMA an nrows x 64-float tile (row stride 64 floats in memory)
// ---- into LDS at byte address ldsAddr, padding 4 dwords after each row.
__device__ __forceinline__ void tdm_load_rows(const float* gsrc,
                                              unsigned ldsAddr, int nrows) {
    unsigned long long ga = (unsigned long long)(uintptr_t)gsrc;
    u32x4 g0;
    g0[0] = 1u;                                   // count=1 (valid), user mode
    g0[1] = ldsAddr;                              // lds_addr  [63:32]
    g0[2] = (unsigned)ga;                         // global_addr lo
    g0[3] = (unsigned)((ga >> 32) & 0x01ffffffu)  // global_addr[56:32]
          | 0x80000000u;                          // type=2 ("image") [127:126]
    i32x8 g1;
    g1[0] = (int)((2u << 16)        // data_size = 4 bytes
                | (1u << 20)        // pad_enable
                | (5u << 22)        // pad_interval: every 64 dwords
                | (3u << 25));      // pad_amount: 4 dwords
    g1[1] = (int)((64u & 0xffffu) << 16);                  // tensor_dim0 lo16
    g1[2] = (int)((64u >> 16) | ((unsigned)nrows << 16));  // dim0 hi | dim1 lo
    g1[3] = (int)(64u << 16);                              // dim1 hi | tile_dim0=64
    g1[4] = (int)(unsigned)nrows;                          // tile_dim1 | tile_dim2=0
    g1[5] = 64;                                            // tensor_dim0_stride lo32
    g1[6] = 0;                                             // stride0 hi | stride1 lo
    g1[7] = 0;
    i32x4 zz4 = {0, 0, 0, 0};
    i32x8 zz8 = {0, 0, 0, 0, 0, 0, 0, 0};
    __builtin_amdgcn_tensor_load_to_lds(g0, g1, zz4, zz4, zz8, 0);
}

__device__ __forceinline__ unsigned lds_off(const void* p) {
    return (unsigned)(uintptr_t)p;   // low 32 bits of flat LDS aperture = offset
}

__launch_bounds__(128)
__global__ void attn_f32wmma_kernel(const float* __restrict__ q,
                                    const float* __restrict__ k,
                                    const float* __restrict__ v,
                                    const unsigned char* __restrict__ mask,
                                    float* __restrict__ out,    // [B,L,D]
                                    float* __restrict__ attn)   // [B,L,L]
{
    __shared__ float qs[16 * QS];
    __shared__ float ks[4][2][16 * QS];   // per-wave double-buffered K tiles
    __shared__ float sc[16 * SS];         // scores -> probabilities
    __shared__ float vs[2][16 * QS];      // double-buffered shared V tile
    __shared__ float pmax[16][8];
    __shared__ float psum[16][8];
    __shared__ float rstat[16];

    const int tid  = threadIdx.x;
    const int lane = tid & 31;
    const int half = lane >> 4;   // lane-half selects K pair / M offset
    const int ln   = lane & 15;
    const int wv   = __builtin_amdgcn_readfirstlane(tid >> 5);  // wave id (SGPR)

    const int b  = blockIdx.x >> 6;   // 64 q-tiles per batch entry
    const int qt = blockIdx.x & 63;
    const int q0 = qt * 16;

    const float* qb = q + ((size_t)b * LL + q0) * DD;
    const float* kb = k + (size_t)b * LL * DD;
    const float* vb = v + (size_t)b * LL * DD;

    // ---- stage Q tile (16 x 64): each wave DMAs its 4-row slice ----
    tdm_load_rows(qb + (size_t)(wv * 4) * DD, lds_off(&qs[wv * 4 * QS]), 4);
    __builtin_amdgcn_s_wait_tensorcnt(0);
    asm volatile("" ::: "memory");
    __syncthreads();

    // ---- preload A(Q) fragments: VGPRj holds Q[m][4*kc + 2*half + j] ----
    v2f aq[16];
#pragma unroll
    for (int kc = 0; kc < 16; ++kc) {
        int kk = kc * 4 + 2 * half;
        aq[kc].x = qs[ln * QS + kk];
        aq[kc].y = qs[ln * QS + kk + 1];
    }

    // ================= pass 1: S = (Q K^T) / T, masked =================
    // wave wv owns key tiles [wv*16, wv*16+16); double-buffered TDM staging
    tdm_load_rows(kb + (size_t)(wv * 256) * DD, lds_off(&ks[wv][0][0]), 16);
    for (int t = 0; t < 16; ++t) {
        const int k0 = (wv * 16 + t) * 16;
        if (t < 15) {
            tdm_load_rows(kb + (size_t)(k0 + 16) * DD,
                          lds_off(&ks[wv][(t + 1) & 1][0]), 16);
            __builtin_amdgcn_s_wait_tensorcnt(1);   // tile t landed
        } else {
            __builtin_amdgcn_s_wait_tensorcnt(0);
        }
        asm volatile("" ::: "memory");
        const float* ksw = ks[wv][t & 1];

        if (t < 15)  // warm the next mask tile rows
            __builtin_prefetch(&mask[((size_t)b * LL + q0 + ln) * LL + k0 + 16], 0, 1);

        v8f c = {};
#pragma unroll
        for (int kc = 0; kc < 16; ++kc) {
            int kk = kc * 4 + 2 * half;
            v2f bf;                       // B[d'][n] = K[k0+n][d']
            bf.x = ksw[ln * QS + kk];
            bf.y = ksw[ln * QS + kk + 1];
            c = __builtin_amdgcn_wmma_f32_16x16x4_f32(
                    false, aq[kc], false, bf, (short)0, c, false, false);
        }

        // scale + mask + scatter into score buffer
#pragma unroll
        for (int r = 0; r < 8; ++r) {
            int m = r + 8 * half;          // C layout: VGPR r -> M = r + 8*half
            float s = c[r] * INV_T;
            unsigned char mv = mask[((size_t)b * LL + (q0 + m)) * LL + (k0 + ln)];
            if (mv) s = -INFINITY;
            sc[m * SS + k0 + ln] = s;
        }
    }
    __syncthreads();

    // ================= softmax over each of 16 rows (len 1024) =========
    const int srow = tid & 15;    // row
    const int sseg = tid >> 4;    // 8 segments of 128 columns
    float mx = -INFINITY;
    for (int c = sseg * 128; c < sseg * 128 + 128; ++c)
        mx = fmaxf(mx, sc[srow * SS + c]);
    pmax[srow][sseg] = mx;
    __syncthreads();
    if (tid < 16) {
        float m2 = pmax[tid][0];
        for (int i = 1; i < 8; ++i) m2 = fmaxf(m2, pmax[tid][i]);
        rstat[tid] = m2;
    }
    __syncthreads();
    const float rm = rstat[srow];
    float sum = 0.f;
    for (int c = sseg * 128; c < sseg * 128 + 128; ++c) {
        float e = __expf(sc[srow * SS + c] - rm);
        sc[srow * SS + c] = e;
        sum += e;
    }
    psum[srow][sseg] = sum;
    __syncthreads();
    if (tid < 16) {
        float s2 = 0.f;
        for (int i = 0; i < 8; ++i) s2 += psum[tid][i];
        rstat[tid] = 1.0f / s2;
    }
    __syncthreads();
    const float rinv = rstat[srow];

    // normalize in LDS and stream probabilities to global attn output
    float* attnRow = attn + ((size_t)b * LL + (q0 + srow)) * LL;
    for (int c4 = sseg * 32; c4 < sseg * 32 + 32; ++c4) {
        v4f p = *(v4f*)(&sc[srow * SS + c4 * 4]);
        p *= rinv;
        *(v4f*)(&sc[srow * SS + c4 * 4]) = p;
        *(v4f*)(&attnRow[c4 * 4]) = p;
    }
    __syncthreads();

    // ================= pass 2: out = P @ V  ============================
    // wave wv produces the 16x16 output tile covering cols [wv*16, wv*16+16).
    // Each wave DMAs its 4-row slice of the shared V tile; double-buffered.
    v8f oc = {};
    tdm_load_rows(vb + (size_t)(wv * 4) * DD, lds_off(&vs[0][wv * 4 * QS]), 4);
    for (int t = 0; t < 64; ++t) {
        if (t < 63) {
            tdm_load_rows(vb + (size_t)((t + 1) * 16 + wv * 4) * DD,
                          lds_off(&vs[(t + 1) & 1][wv * 4 * QS]), 4);
            __builtin_amdgcn_s_wait_tensorcnt(1);   // own slice of tile t landed
        } else {
            __builtin_amdgcn_s_wait_tensorcnt(0);
        }
        asm volatile("" ::: "memory");
        __syncthreads();           // all waves' slices of tile t landed
        const float* vsw = vs[t & 1];
#pragma unroll
        for (int kc = 0; kc < 4; ++kc) {
            int kk = t * 16 + kc * 4 + 2 * half;  // global K for A (probs)
            int kl = kc * 4 + 2 * half;           // local row in V tile
            v2f af, bf;
            af.x = sc[ln * SS + kk];
            af.y = sc[ln * SS + kk + 1];
            bf.x = vsw[kl * QS + wv * 16 + ln];
            bf.y = vsw[(kl + 1) * QS + wv * 16 + ln];
            oc = __builtin_amdgcn_wmma_f32_16x16x4_f32(
                     false, af, false, bf, (short)0, oc, false, false);
        }
        __syncthreads();           // everyone done reading before overwrite
    }

    // write the 16x16 output tile
    float* ob = out + ((size_t)b * LL + q0) * DD + wv * 16;
#pragma unroll
    for (int r = 0; r < 8; ++r) {
        int m = r + 8 * half;
        ob[(size_t)m * DD + ln] = oc[r];
    }
}

extern "C" void kernel_launch(void* const* d_in, const int* in_sizes, int n_in,
                              void* d_out, int out_size, void* d_ws, size_t ws_size,
                              hipStream_t stream) {
    (void)in_sizes; (void)n_in; (void)out_size; (void)d_ws; (void)ws_size;
    const float* q = (const float*)d_in[0];
    const float* k = (const float*)d_in[1];
    const float* v = (const float*)d_in[2];
    const unsigned char* mask = (const unsigned char*)d_in[3];
    float* out  = (float*)d_out;                       // [B,L,D] first
    float* attn = out + (size_t)BB * LL * DD;          // then [B,L,L]

    attn_f32wmma_kernel<<<dim3(BB * (LL / 16)), dim3(128), 0, stream>>>(
        q, k, v, mask, out, attn);
}